// ResGINLayer_26242250178930
// MI455X (gfx1250) — compile-verified
//
#include <hip/hip_runtime.h>

// ---------------------------------------------------------------------------
// ResGIN layer for MI455X (gfx1250, wave32, WMMA f32 16x16x4).
//   agg = segment_sum(x[src], dst)
//   h   = (1+eps)*x + agg
//   h1  = relu(h @ W1 + b1)
//   out = BN(h1 @ W2 + b2) + x @ Wp + bp
//
// GEMM tiling: 32-row M blocks (N=100000=3125*32 exact), block=256thr=8 waves.
// Wave w owns N-cols [32w,32w+32) as two 16-wide tiles, and BOTH 16-row M
// tiles of the block (2x2 accumulator grid) so each B fragment feeds 2 WMMAs.
// ---------------------------------------------------------------------------

#define IN_DIM  128
#define OD      256
#define BN_EPS  1e-5f
#define MROWS   32          // M rows per block (two 16-row WMMA tiles)

typedef float v2f __attribute__((ext_vector_type(2)));
typedef float v8f __attribute__((ext_vector_type(8)));

// ---------------------------------------------------------------------------
// Zero-fill (float4 granularity)
// ---------------------------------------------------------------------------
__global__ void resgin_zero_kernel(float* __restrict__ p, int n4) {
    int i = blockIdx.x * blockDim.x + threadIdx.x;
    if (i < n4) ((float4*)p)[i] = make_float4(0.f, 0.f, 0.f, 0.f);
}

// ---------------------------------------------------------------------------
// Scatter-add aggregation: agg[dst[e]] += x[src[e]]
// 32 threads per edge, float4 gather (512B coalesced per edge) + f32 atomics.
// ---------------------------------------------------------------------------
__global__ void resgin_scatter_kernel(const float* __restrict__ x,
                                      const long long* __restrict__ edge_index,
                                      float* __restrict__ agg, int E) {
    int t = blockIdx.x * blockDim.x + threadIdx.x;
    if (t >= E * 32) return;
    int e = t >> 5;
    int q = (t & 31) << 2;                      // feature offset 0..124 step 4
    long long s = edge_index[e];                // src row
    long long d = edge_index[E + e];            // dst row
    const float4 v = *(const float4*)(x + s * IN_DIM + q);
    float* p = agg + d * IN_DIM + q;
    unsafeAtomicAdd(p + 0, v.x);
    unsafeAtomicAdd(p + 1, v.y);
    unsafeAtomicAdd(p + 2, v.z);
    unsafeAtomicAdd(p + 3, v.w);
}

// ---------------------------------------------------------------------------
// WMMA core: 2x2 tile grid per wave.  A (32 x K, row-major) in LDS,
// B (K x 256) in global (weights, L2-resident).
//
// Fragment layouts per CDNA5 ISA 7.12.2 (V_WMMA_F32_16X16X4_F32):
//   A 16x4 f32 : row = lane&15; lanes 0-15 -> v0=K0,v1=K1; lanes 16-31 -> K2,K3
//   B 4x16 f32 : col = lane&15; lanes 0-15 -> v0=K0,v1=K1; lanes 16-31 -> K2,K3
//   C/D 16x16  : VGPR r -> row r + 8*(lane>>4), col lane&15
// ---------------------------------------------------------------------------
template <int K>
__device__ __forceinline__ void resgin_wmma_core(const float* __restrict__ lA,
                                                 const float* __restrict__ B,
                                                 int lane, int n0,
                                                 v8f& c00, v8f& c01,
                                                 v8f& c10, v8f& c11) {
    const int t  = lane & 15;
    const int kh = (lane >> 4) << 1;            // 0 for lanes 0-15, 2 for 16-31
    #pragma unroll 4
    for (int k0 = 0; k0 < K; k0 += 4) {
        const float* ar0 = lA + t * K + k0 + kh;        // M tile 0
        const float* ar1 = ar0 + 16 * K;                // M tile 1
        const float* br  = B + (size_t)(k0 + kh) * OD + n0 + t;
        v2f a0, a1, b0, b1;
        a0[0] = ar0[0];   a0[1] = ar0[1];
        a1[0] = ar1[0];   a1[1] = ar1[1];
        b0[0] = br[0];    b0[1] = br[OD];               // N tile 0
        b1[0] = br[16];   b1[1] = br[OD + 16];          // N tile 1
        c00 = __builtin_amdgcn_wmma_f32_16x16x4_f32(false, a0, false, b0,
                                                    (short)0, c00, false, false);
        c01 = __builtin_amdgcn_wmma_f32_16x16x4_f32(false, a0, false, b1,
                                                    (short)0, c01, false, false);
        c10 = __builtin_amdgcn_wmma_f32_16x16x4_f32(false, a1, false, b0,
                                                    (short)0, c10, false, false);
        c11 = __builtin_amdgcn_wmma_f32_16x16x4_f32(false, a1, false, b1,
                                                    (short)0, c11, false, false);
    }
}

// ---------------------------------------------------------------------------
// GEMM1 (fused GIN combine): h1 = relu( ((1+eps)*x + agg) @ W1 + b1 )
// ---------------------------------------------------------------------------
__global__ void __launch_bounds__(256)
resgin_mlp1_kernel(const float* __restrict__ x, const float* __restrict__ agg,
                   const float* __restrict__ eps_p,
                   const float* __restrict__ W1, const float* __restrict__ b1,
                   float* __restrict__ h1) {
    __shared__ float lA[MROWS * IN_DIM];
    const int m0 = blockIdx.x * MROWS;
    const float e1 = 1.0f + *eps_p;

    for (int i = threadIdx.x * 4; i < MROWS * IN_DIM; i += blockDim.x * 4) {
        int row = i / IN_DIM, col = i % IN_DIM;
        const float4 xv = *(const float4*)(x   + (size_t)(m0 + row) * IN_DIM + col);
        const float4 av = *(const float4*)(agg + (size_t)(m0 + row) * IN_DIM + col);
        *(float4*)(lA + i) = make_float4(e1 * xv.x + av.x, e1 * xv.y + av.y,
                                         e1 * xv.z + av.z, e1 * xv.w + av.w);
    }
    __syncthreads();

    const int lane = threadIdx.x & 31;
    const int n0   = (threadIdx.x >> 5) * 32;
    v8f c00 = {}, c01 = {}, c10 = {}, c11 = {};
    resgin_wmma_core<IN_DIM>(lA, W1, lane, n0, c00, c01, c10, c11);

    const int t  = lane & 15;
    const int mo = (lane >> 4) * 8;
    const float ba = b1[n0 + t], bb = b1[n0 + 16 + t];
    #pragma unroll
    for (int r = 0; r < 8; ++r) {
        const size_t ma = (size_t)(m0 + mo + r);
        const size_t mb = ma + 16;
        float v00 = c00[r] + ba, v01 = c01[r] + bb;
        float v10 = c10[r] + ba, v11 = c11[r] + bb;
        h1[ma * OD + n0 + t]      = v00 > 0.f ? v00 : 0.f;
        h1[ma * OD + n0 + 16 + t] = v01 > 0.f ? v01 : 0.f;
        h1[mb * OD + n0 + t]      = v10 > 0.f ? v10 : 0.f;
        h1[mb * OD + n0 + 16 + t] = v11 > 0.f ? v11 : 0.f;
    }
}

// ---------------------------------------------------------------------------
// Projection GEMM: out = x @ Wp + bp   (overwrites d_out)
// ---------------------------------------------------------------------------
__global__ void __launch_bounds__(256)
resgin_proj_kernel(const float* __restrict__ x,
                   const float* __restrict__ Wp, const float* __restrict__ bp,
                   float* __restrict__ out) {
    __shared__ float lA[MROWS * IN_DIM];
    const int m0 = blockIdx.x * MROWS;

    for (int i = threadIdx.x * 4; i < MROWS * IN_DIM; i += blockDim.x * 4) {
        int row = i / IN_DIM, col = i % IN_DIM;
        *(float4*)(lA + i) = *(const float4*)(x + (size_t)(m0 + row) * IN_DIM + col);
    }
    __syncthreads();

    const int lane = threadIdx.x & 31;
    const int n0   = (threadIdx.x >> 5) * 32;
    v8f c00 = {}, c01 = {}, c10 = {}, c11 = {};
    resgin_wmma_core<IN_DIM>(lA, Wp, lane, n0, c00, c01, c10, c11);

    const int t  = lane & 15;
    const int mo = (lane >> 4) * 8;
    const float ba = bp[n0 + t], bb = bp[n0 + 16 + t];
    #pragma unroll
    for (int r = 0; r < 8; ++r) {
        const size_t ma = (size_t)(m0 + mo + r);
        const size_t mb = ma + 16;
        out[ma * OD + n0 + t]      = c00[r] + ba;
        out[ma * OD + n0 + 16 + t] = c01[r] + bb;
        out[mb * OD + n0 + t]      = c10[r] + ba;
        out[mb * OD + n0 + 16 + t] = c11[r] + bb;
    }
}

// ---------------------------------------------------------------------------
// GEMM2 + BatchNorm epilogue, accumulated into d_out:  out += BN(h1 @ W2 + b2)
// ---------------------------------------------------------------------------
__global__ void __launch_bounds__(256)
resgin_mlp2_kernel(const float* __restrict__ h1,
                   const float* __restrict__ W2, const float* __restrict__ b2,
                   const float* __restrict__ gamma, const float* __restrict__ beta,
                   const float* __restrict__ run_mean, const float* __restrict__ run_var,
                   float* __restrict__ out) {
    __shared__ float lA[MROWS * OD];
    const int m0 = blockIdx.x * MROWS;

    for (int i = threadIdx.x * 4; i < MROWS * OD; i += blockDim.x * 4) {
        int row = i / OD, col = i % OD;
        *(float4*)(lA + i) = *(const float4*)(h1 + (size_t)(m0 + row) * OD + col);
    }
    __syncthreads();

    const int lane = threadIdx.x & 31;
    const int n0   = (threadIdx.x >> 5) * 32;
    v8f c00 = {}, c01 = {}, c10 = {}, c11 = {};
    resgin_wmma_core<OD>(lA, W2, lane, n0, c00, c01, c10, c11);

    const int t  = lane & 15;
    const int mo = (lane >> 4) * 8;
    const int na = n0 + t, nb = n0 + 16 + t;
    const float sa = rsqrtf(run_var[na] + BN_EPS) * gamma[na];
    const float sb = rsqrtf(run_var[nb] + BN_EPS) * gamma[nb];
    const float oa = beta[na] + (b2[na] - run_mean[na]) * sa;
    const float ob = beta[nb] + (b2[nb] - run_mean[nb]) * sb;
    #pragma unroll
    for (int r = 0; r < 8; ++r) {
        const size_t ma = (size_t)(m0 + mo + r);
        const size_t mb = ma + 16;
        out[ma * OD + na] += c00[r] * sa + oa;
        out[ma * OD + nb] += c01[r] * sb + ob;
        out[mb * OD + na] += c10[r] * sa + oa;
        out[mb * OD + nb] += c11[r] * sb + ob;
    }
}

// ---------------------------------------------------------------------------
// Host launch
// ---------------------------------------------------------------------------
extern "C" void kernel_launch(void* const* d_in, const int* in_sizes, int n_in,
                              void* d_out, int out_size, void* d_ws, size_t ws_size,
                              hipStream_t stream) {
    const float*     x   = (const float*)d_in[0];
    const long long* ei  = (const long long*)d_in[1];
    const float*     eps = (const float*)d_in[2];
    const float*     W1  = (const float*)d_in[3];
    const float*     b1  = (const float*)d_in[4];
    const float*     W2  = (const float*)d_in[5];
    const float*     b2  = (const float*)d_in[6];
    const float*     gam = (const float*)d_in[7];
    const float*     bet = (const float*)d_in[8];
    const float*     rmu = (const float*)d_in[9];
    const float*     rvr = (const float*)d_in[10];
    const float*     Wp  = (const float*)d_in[11];
    const float*     bp  = (const float*)d_in[12];
    float*           out = (float*)d_out;

    const int N = in_sizes[0] / IN_DIM;      // 100000
    const int E = in_sizes[1] / 2;           // 600000

    float* agg = (float*)d_ws;                       // N*128 floats
    float* h1  = agg + (size_t)N * IN_DIM;           // N*256 floats

    // 1) agg = 0
    {
        int n4 = (N * IN_DIM) / 4;
        resgin_zero_kernel<<<(n4 + 255) / 256, 256, 0, stream>>>(agg, n4);
    }
    // 2) agg[dst] += x[src]
    {
        long long threads = (long long)E * 32;
        int blocks = (int)((threads + 255) / 256);
        resgin_scatter_kernel<<<blocks, 256, 0, stream>>>(x, ei, agg, E);
    }
    const int mblocks = N / MROWS;           // 3125 (exact)
    // 3) h1 = relu(((1+eps)x + agg) @ W1 + b1)
    resgin_mlp1_kernel<<<mblocks, 256, 0, stream>>>(x, agg, eps, W1, b1, h1);
    // 4) out = x @ Wp + bp
    resgin_proj_kernel<<<mblocks, 256, 0, stream>>>(x, Wp, bp, out);
    // 5) out += BN(h1 @ W2 + b2)
    resgin_mlp2_kernel<<<mblocks, 256, 0, stream>>>(h1, W2, b2, gam, bet, rmu, rvr, out);
}